// CapsuleLayer_50603304681967
// MI455X (gfx1250) — compile-verified
//
#include <hip/hip_runtime.h>
#include <stdint.h>

// ---------------- problem constants (from reference setup_inputs) ------------
#define B_TOT   64
#define R_TOT   2048
#define CIN     16
#define C_TOT   32
#define O_TOT   32

#define BG      16                 // b's per block (WMMA M dimension)
#define RC      16                 // r's per chunk
#define NCHUNK  (R_TOT / RC)       // 128
#define THREADS 256
#define NWAVE   (THREADS / 32)     // 8 waves (wave32)

typedef float v2f __attribute__((ext_vector_type(2)));
typedef float v8f __attribute__((ext_vector_type(8)));

// D(16x16 f32) = A(16x4 f32) * B(4x16 f32) + C
__device__ __forceinline__ v8f wmma_f32_16x16x4(v2f a, v2f b, v8f c) {
  return __builtin_amdgcn_wmma_f32_16x16x4_f32(
      /*neg_a=*/false, a, /*neg_b=*/false, b,
      /*c_mod=*/(short)0, c, /*reuse_a=*/false, /*reuse_b=*/false);
}

// Async memory -> LDS copy, 16B per lane, tracked by ASYNCcnt.
// VDST operand carries the per-lane LDS byte offset (low 32 bits of a generic
// LDS pointer are the wave-relative LDS offset).
__device__ __forceinline__ void async_b128(float* lds_dst, const float* gsrc) {
  uint32_t ldso = (uint32_t)(uintptr_t)lds_dst;
  asm volatile("global_load_async_to_lds_b128 %0, %1, off"
               :: "v"(ldso), "v"(gsrc)
               : "memory");
}
#define S_WAIT_ASYNC_0()  asm volatile("s_wait_asynccnt 0x0" ::: "memory")
#define S_WAIT_ASYNC_12() asm volatile("s_wait_asynccnt 0xc" ::: "memory")

// Issue the 12 async b128 ops that stage one (X,W) chunk into LDS.
__device__ __forceinline__ void stage_chunk_async(const float* __restrict__ xbase,
                                                  const float* __restrict__ wbase,
                                                  int r0, float* Xd, float* Wd,
                                                  int tid) {
#pragma unroll
  for (int j = 0; j < 4; ++j) {                 // X: [BG][RC][CIN] = 16KB
    int q  = tid + j * THREADS;                 // float4 index
    int L  = q * 4;
    int bi = L >> 8;                            // which b (256 floats per b)
    int re = L & 255;                           // (rr,i) linear, contiguous
    async_b128(&Xd[L],
               &xbase[(size_t)bi * (R_TOT * CIN) + (size_t)r0 * CIN + re]);
  }
  const float* wsrc = wbase + (size_t)r0 * (CIN * O_TOT);
#pragma unroll
  for (int j = 0; j < 8; ++j) {                 // W: [RC][CIN][O] = 32KB contig
    int q = (tid + j * THREADS) * 4;
    async_b128(&Wd[q], &wsrc[q]);
  }
}

__global__ void __launch_bounds__(THREADS)
capsule_route_fused(const float* __restrict__ x,     // [B,R,CIN]
                    const float* __restrict__ w,     // [C,R,CIN,O]
                    float* __restrict__ out)         // [C,B,1,1,O]
{
  extern __shared__ float smem[];
  float* Xs0 = smem;                         // [BG][RC][CIN]   4096 f
  float* Xs1 = Xs0 + BG * RC * CIN;          //                 4096 f
  float* Ws0 = Xs1 + BG * RC * CIN;          // [RC][CIN][O]    8192 f
  float* Ws1 = Ws0 + RC * CIN * O_TOT;       //                 8192 f
  float* Ps0 = Ws1 + RC * CIN * O_TOT;       // [BG][RC][O]     8192 f
  float* Ps1 = Ps0 + BG * RC * O_TOT;        //                 8192 f
  float* Sb  = Ps1 + BG * RC * O_TOT;        // [256][O]        8192 f
  float* mb  = Sb + THREADS * O_TOT;         // [256]
  float* db  = mb + THREADS;                 // [256]
  float* us  = db + THREADS;                 // [BG][O] routing vector

  const int tid  = threadIdx.x;
  const int wave = tid >> 5;
  const int lane = tid & 31;
  const int hi   = lane >> 4;                // 0/1: which half-wave
  const int ln16 = lane & 15;

  const int c  = blockIdx.x >> 2;            // capsule class
  const int b0 = (blockIdx.x & 3) * BG;      // b-group base

  const int bb_t = tid >> 4;                 // routing: this thread's b
  const int rr_t = tid & 15;                 // routing: r-within-chunk

  const float* xbase = x + (size_t)b0 * R_TOT * CIN;
  const float* wbase = w + (size_t)c * R_TOT * CIN * O_TOT;

  for (int pass = 0; pass < 3; ++pass) {
    float accS[O_TOT];
#pragma unroll
    for (int o = 0; o < O_TOT; ++o) accS[o] = 0.0f;
    float m_run = -3.0e38f, d_run = 0.0f;

    // pipeline prologue: stage chunk 0 into buffer 0
    stage_chunk_async(xbase, wbase, 0, Xs0, Ws0, tid);

    for (int ch = 0; ch < NCHUNK; ++ch) {
      float* Xc = (ch & 1) ? Xs1 : Xs0;
      float* Wc = (ch & 1) ? Ws1 : Ws0;
      float* Pc = (ch & 1) ? Ps1 : Ps0;

      // stage next chunk into the other buffer, then wait until only those
      // 12 async ops remain outstanding => this chunk's data has landed
      if (ch + 1 < NCHUNK) {
        stage_chunk_async(xbase, wbase, (ch + 1) * RC,
                          (ch & 1) ? Xs0 : Xs1, (ch & 1) ? Ws0 : Ws1, tid);
        S_WAIT_ASYNC_12();
      } else {
        S_WAIT_ASYNC_0();
      }
      __syncthreads();   // barrier A: staged X/W visible to all waves

      // ---- WMMA: 16 r x 2 n-tiles = 32 groups, 4 per wave ----------------
#pragma unroll
      for (int j = 0; j < 4; ++j) {
        int g  = wave + j * NWAVE;            // 0..31
        int rr = g & 15;
        int nt = g >> 4;
        int oo = nt * 16 + ln16;              // B/D column (o)
        const float* xrow = &Xc[ln16 * (RC * CIN) + rr * CIN + hi * 2];
        const float* wrow = &Wc[rr * (CIN * O_TOT) + (hi * 2) * O_TOT + oo];
        v2f a[4], bf[4];
#pragma unroll
        for (int kk = 0; kk < 4; ++kk) {      // preload all fragments
          a[kk]    = *(const v2f*)&xrow[kk * 4];
          bf[kk].x = wrow[(kk * 4 + 0) * O_TOT];
          bf[kk].y = wrow[(kk * 4 + 1) * O_TOT];
        }
        v8f acc = {};
#pragma unroll
        for (int kk = 0; kk < 4; ++kk)        // back-to-back WMMA chain
          acc = wmma_f32_16x16x4(a[kk], bf[kk], acc);
#pragma unroll
        for (int v = 0; v < 8; ++v) {         // D tile -> priors chunk
          int bbo = v + hi * 8;
          Pc[bbo * (RC * O_TOT) + rr * O_TOT + oo] = acc[v];
        }
      }
      __syncthreads();   // barrier B: priors chunk visible

      // ---- routing math: thread owns (bb_t, rr_t) row of the chunk -------
      const float* row = &Pc[bb_t * (RC * O_TOT) + rr_t * O_TOT];
      if (pass == 0) {
#pragma unroll
        for (int o = 0; o < O_TOT; o += 4) {
          float4 pv = *(const float4*)&row[o];
          accS[o + 0] += pv.x; accS[o + 1] += pv.y;
          accS[o + 2] += pv.z; accS[o + 3] += pv.w;
        }
      } else {
        // logit_r = prior_r . u   (u = v0 for pass1, v0+v1 for pass2)
        const float* uv = &us[bb_t * O_TOT];
        float t = 0.0f;
#pragma unroll
        for (int o = 0; o < O_TOT; ++o) t += row[o] * uv[o];
        float nm = fmaxf(m_run, t);
        float sc = __expf(m_run - nm);
        float p  = __expf(t - nm);
        d_run = d_run * sc + p;
#pragma unroll
        for (int o = 0; o < O_TOT; ++o) accS[o] = accS[o] * sc + p * row[o];
        m_run = nm;
      }
      // no barrier needed: Ps is double-buffered; next iteration's barrier A
      // orders the next WMMA writes after this routing read
    } // chunks

    // ---- publish per-thread partials, merge 16 threads per b -------------
    mb[tid] = m_run;
    db[tid] = d_run;
#pragma unroll
    for (int o = 0; o < O_TOT; ++o) Sb[tid * O_TOT + o] = accS[o];
    __syncthreads();

    if (tid < BG) {                           // leader per b
      float s[O_TOT];
      if (pass == 0) {
#pragma unroll
        for (int o = 0; o < O_TOT; ++o) s[o] = 0.0f;
        for (int r2 = 0; r2 < 16; ++r2) {
          const float* pr = &Sb[(tid * 16 + r2) * O_TOT];
#pragma unroll
          for (int o = 0; o < O_TOT; ++o) s[o] += pr[o];
        }
        const float invR = 1.0f / (float)R_TOT;
#pragma unroll
        for (int o = 0; o < O_TOT; ++o) s[o] *= invR;
      } else {
        float M = -3.0e38f;
        for (int r2 = 0; r2 < 16; ++r2) M = fmaxf(M, mb[tid * 16 + r2]);
        float dt = 0.0f;
#pragma unroll
        for (int o = 0; o < O_TOT; ++o) s[o] = 0.0f;
        for (int r2 = 0; r2 < 16; ++r2) {
          float w8 = __expf(mb[tid * 16 + r2] - M);
          dt += db[tid * 16 + r2] * w8;
          const float* pr = &Sb[(tid * 16 + r2) * O_TOT];
#pragma unroll
          for (int o = 0; o < O_TOT; ++o) s[o] += pr[o] * w8;
        }
        float inv = 1.0f / dt;
#pragma unroll
        for (int o = 0; o < O_TOT; ++o) s[o] *= inv;
      }
      // squash: scale = |s|^2 / (|s|^2 + 1);  out = scale * s / |s|
      float n2 = 0.0f;
#pragma unroll
      for (int o = 0; o < O_TOT; ++o) n2 += s[o] * s[o];
      float l2   = sqrtf(n2);
      float coef = (n2 / (n2 + 1.0f)) / l2;
      if (pass == 0) {
#pragma unroll
        for (int o = 0; o < O_TOT; ++o) us[tid * O_TOT + o] = s[o] * coef;
      } else if (pass == 1) {
#pragma unroll
        for (int o = 0; o < O_TOT; ++o) us[tid * O_TOT + o] += s[o] * coef;
      } else {
        float* op = out + ((size_t)c * B_TOT + (b0 + tid)) * O_TOT;
#pragma unroll
        for (int o = 0; o < O_TOT; ++o) op[o] = s[o] * coef;
      }
    }
    __syncthreads();
  } // passes
}

extern "C" void kernel_launch(void* const* d_in, const int* in_sizes, int n_in,
                              void* d_out, int out_size, void* d_ws, size_t ws_size,
                              hipStream_t stream) {
  (void)in_sizes; (void)n_in; (void)d_ws; (void)ws_size; (void)out_size;
  const float* x = (const float*)d_in[0];              // [64,2048,16]
  const float* w = (const float*)d_in[1];              // [32,2048,16,32]
  float* out = (float*)d_out;                          // [32,64,1,1,32]

  const size_t smem_floats =
      2 * (size_t)BG * RC * CIN +     // Xs0/Xs1    8192
      2 * (size_t)RC * CIN * O_TOT +  // Ws0/Ws1   16384
      2 * (size_t)BG * RC * O_TOT +   // Ps0/Ps1   16384
      (size_t)THREADS * O_TOT +       // Sb         8192
      THREADS + THREADS +             // mb, db
      (size_t)BG * O_TOT;             // us
  const size_t smem = smem_floats * sizeof(float);     // ~196 KB of 320 KB

  dim3 grid(C_TOT * (B_TOT / BG));                     // 128 blocks
  capsule_route_fused<<<grid, THREADS, smem, stream>>>(x, w, out);
}